// NGramsEmbedding_11072425689872
// MI455X (gfx1250) — compile-verified
//
#include <hip/hip_runtime.h>

typedef __attribute__((ext_vector_type(2))) float v2f;
typedef __attribute__((ext_vector_type(8))) float v8f;

#define VOCAB 32000
#define EMBED 256
#define NPOS  4096   // 128 * 32 positions
#define NGRAM 3

// Pack per-K-slot constants: off = (K%3)*16 + K/3 indexes the LDS row/flag
// tables; q = K/3 is the position this K column belongs to.
//   bits [ 5: 0] off0   (slot K)
//   bits [11: 6] off1   (slot K+1)
//   bits [15:12] q0
//   bits [19:16] q1
__host__ __device__ constexpr unsigned mkcfg(unsigned K) {
    return ((K % 3u) * 16u + K / 3u)
         | ((((K + 1u) % 3u) * 16u + (K + 1u) / 3u) << 6)
         | ((K / 3u) << 12)
         | (((K + 1u) / 3u) << 16);
}

// One wave32 computes a 16(pos) x 32(dim) output tile as
//   D = A(16x48) * B(48x32)
// A = 0/1 n-hot selection matrix (dedupe flags folded in, exact in f32),
// B = 48 gathered weight rows. 12 K-groups x 2 N-tiles = 24
// v_wmma_f32_16x16x4_f32 per wave. Each weight element is gathered exactly
// once per 16-position group -> traffic matches the scalar-gather roofline
// (~12 MB reads + 4 MB NT writes). Branchless; EXEC untouched in the loop.
__global__ __launch_bounds__(256) void ngrams_embed_wmma(
    const int* __restrict__ inp,      // [3, 4096]
    const float* __restrict__ weight, // [32000, 256]
    float* __restrict__ out)          // [4096, 256]
{
    // sTab[ 0..47]: row id for K-slot off = t*16+q
    // sTab[48..95]: f32 flag (1.0f if gram t at position q counts, else 0.0f)
    __shared__ int sTab[96];

    const int pg  = blockIdx.x;       // position group (16 positions)
    const int p0  = pg * 16;
    const int tid = threadIdx.x;

    if (tid < 16) {
        const int q  = tid;
        const int i0 = inp[p0 + q];
        const int i1 = inp[NPOS + p0 + q];
        const int i2 = inp[2 * NPOS + p0 + q];
        sTab[q]      = i0;
        sTab[16 + q] = i1;
        sTab[32 + q] = i2;
        // n_hot SET semantics: duplicates count once. Flags stored as f32 bits.
        sTab[48 + q] = __float_as_int(1.0f);
        sTab[64 + q] = (i1 != i0) ? __float_as_int(1.0f) : 0;
        sTab[80 + q] = ((i2 != i0) && (i2 != i1)) ? __float_as_int(1.0f) : 0;
    }
    __syncthreads();

    const int wid  = tid >> 5;        // 0..7 -> 32-dim chunk of EMBED
    const int lane = tid & 31;
    const int half = lane >> 4;       // 0: lanes 0-15, 1: lanes 16-31
    const unsigned r = lane & 15;     // M index for A, N index for B/D
    const unsigned colbase = (unsigned)wid * 32u + r;

    v8f acc0 = {};                    // N-tile 0 (cols colbase + 0)
    v8f acc1 = {};                    // N-tile 1 (cols colbase + 16)

#pragma unroll
    for (int kg = 0; kg < 12; ++kg) {
        // Per ISA A-layout (16x4 f32): lanes 0-15 hold K = 4kg+{0,1},
        // lanes 16-31 hold K = 4kg+{2,3}. B (4x16) mirrors this per VGPR.
        // One cndmask between two opaque literals; fields via bfe/and.
        const unsigned cfg = half ? mkcfg(4u * kg + 2u) : mkcfg(4u * kg);
        const unsigned off0 = cfg & 63u;
        const unsigned off1 = (cfg >> 6) & 63u;
        const unsigned q0   = (cfg >> 12) & 15u;
        const unsigned q1   = (cfg >> 16) & 15u;

        // Row id + f32 flag share the dynamic offset (192B apart) ->
        // ds_load_2addr_b32 candidates.
        const unsigned row0 = (unsigned)sTab[off0];
        const float    fl0  = __int_as_float(sTab[48 + off0]);
        const unsigned row1 = (unsigned)sTab[off1];
        const float    fl1  = __int_as_float(sTab[48 + off1]);

        v2f a;
        a.x = (r == q0) ? fl0 : 0.0f;
        a.y = (r == q1) ? fl1 : 0.0f;

        // 32-bit unsigned voffset (max 32000*256 < 2^23) -> GVS addressing.
        const float* b0p = weight + ((row0 << 8) + colbase);
        const float* b1p = weight + ((row1 << 8) + colbase);

        v2f b;
        b.x = b0p[0];
        b.y = b1p[0];
        acc0 = __builtin_amdgcn_wmma_f32_16x16x4_f32(
            false, a, false, b, (short)0, acc0, false, false);

        b.x = b0p[16];
        b.y = b1p[16];
        acc1 = __builtin_amdgcn_wmma_f32_16x16x4_f32(
            false, a, false, b, (short)0, acc1, false, false);
    }

    // D layout (16x16 f32): VGPR j -> M=j (lanes 0-15), M=j+8 (lanes 16-31), N=r
    // Output is write-once / no-reuse: non-temporal stores, 32-bit voffsets.
#pragma unroll
    for (int j = 0; j < 8; ++j) {
        const unsigned m    = (unsigned)j + 8u * (unsigned)half;
        const unsigned ooff = (((unsigned)p0 + m) << 8) + (unsigned)wid * 32u + r;
        __builtin_nontemporal_store(acc0[j], out + ooff);
        __builtin_nontemporal_store(acc1[j], out + ooff + 16u);
    }
}

extern "C" void kernel_launch(void* const* d_in, const int* in_sizes, int n_in,
                              void* d_out, int out_size, void* d_ws, size_t ws_size,
                              hipStream_t stream) {
    const int*   inp    = (const int*)d_in[0];    // [3, 128, 32] indices
    const float* weight = (const float*)d_in[1];  // [32000, 256]
    float*       outp   = (float*)d_out;          // [128, 32, 256]

    // 256 blocks x 256 threads: one block per 16-position group,
    // 8 waves per block each owning a 32-wide slice of EMBED.
    ngrams_embed_wmma<<<dim3(NPOS / 16), dim3(256), 0, stream>>>(inp, weight, outp);
}